// Attention_39230231281735
// MI455X (gfx1250) — compile-verified
//
#include <hip/hip_runtime.h>
#include <hip/hip_bf16.h>

// CDNA5 / gfx1250, wave32. Point-transformer attention:
//  1) qkv = feats @ qkv_w^T + qkv_b     (WMMA f32 16x16x4, 1x4 register-blocked)
//  2) neighborhood attention (K=16, H=8, HD=32) (VALU + shfl, L2-resident gathers)
//  3) out = x @ proj_w^T + proj_b       (WMMA f32 16x16x4, 1x4 register-blocked)

#if defined(__HIP_DEVICE_COMPILE__) && !__has_builtin(__builtin_amdgcn_wmma_f32_16x16x4_f32)
#error "device pass: missing __builtin_amdgcn_wmma_f32_16x16x4_f32"
#endif

typedef __attribute__((ext_vector_type(2))) float v2f;
typedef __attribute__((ext_vector_type(8))) float v8f;

#define NPTS 50000
#define KNBR 16
#define DIM  256
#define NH   8
#define HD   32
#define NBLK 4                           // N-tiles per wave (A-fragment reuse)
#define QK_SCALE 0.1767766952966369f     // 1/sqrt(32)

// D = A(16x4 f32) * B(4x16 f32) + C ; guarded so the host pass never sees the
// amdgcn builtin (it is only declared when targeting amdgcn).
__device__ inline v8f wmma_f32_16x16x4(v2f a, v2f b, v8f c) {
#if defined(__HIP_DEVICE_COMPILE__)
    return __builtin_amdgcn_wmma_f32_16x16x4_f32(
        false, a, false, b, (short)0, c, false, false);
#else
    (void)a; (void)b;
    return c;
#endif
}

// ---------------------------------------------------------------------------
// Fused QKV projection. One wave per (16 x 64) output strip: 1 M-tile x 4
// N-tiles, so each A fragment is loaded once and feeds 4 WMMAs (A is the
// 51 MB streaming operand; B = weights stay hot in WGP$/L2).
// A fragment (16x4 f32): lane<16 holds row=lane, K={k0,k0+1}; lane>=16 holds
// row=lane-16, K={k0+2,k0+3}.  B fragment (4x16) mirrors with cols.
// C/D: vgpr r, lane l -> M = r + 8*(l>>4), N = l&15.
// ---------------------------------------------------------------------------
__global__ __launch_bounds__(128) void qkv_kernel(
    const float* __restrict__ feats,
    const float* __restrict__ qkv_w,   // (768, 256) row-major
    const float* __restrict__ qkv_b,   // (768,)
    float* __restrict__ q,             // (N, 256)  pre-scaled
    float* __restrict__ k,             // (N, 256)
    float* __restrict__ v)             // (N, 256)
{
    const int GROUPS_N = (3 * DIM / 16) / NBLK;     // 12 groups of 4 col-tiles
    const int wave = (blockIdx.x * blockDim.x + threadIdx.x) >> 5;
    const int lane = threadIdx.x & 31;
    const int tm = wave / GROUPS_N;                 // 0..3124
    const int tn0 = (wave % GROUPS_N) * NBLK;       // first col-tile of group
    const int row  = lane & 15;
    const int half = lane >> 4;

    const float* __restrict__ arow = feats + (size_t)(tm * 16 + row) * DIM + half * 2;
    const float* __restrict__ brow[NBLK];
#pragma unroll
    for (int t = 0; t < NBLK; ++t)
        brow[t] = qkv_w + (size_t)((tn0 + t) * 16 + row) * DIM + half * 2;

    v8f acc[NBLK];
#pragma unroll
    for (int t = 0; t < NBLK; ++t) acc[t] = (v8f){0.f,0.f,0.f,0.f,0.f,0.f,0.f,0.f};

#pragma unroll 4
    for (int k0 = 0; k0 < DIM; k0 += 4) {
        v2f a = *(const v2f*)(arow + k0);
#pragma unroll
        for (int t = 0; t < NBLK; ++t) {
            v2f b = *(const v2f*)(brow[t] + k0);
            acc[t] = wmma_f32_16x16x4(a, b, acc[t]);
        }
    }

#pragma unroll
    for (int t = 0; t < NBLK; ++t) {
        const int col = (tn0 + t) * 16 + row;       // 0..767
        const float bias = qkv_b[col];
        const int part = col >> 8;                  // 0=q 1=k 2=v (wave-uniform)
        const int c = col & (DIM - 1);
        float* __restrict__ dst = (part == 0) ? q : (part == 1) ? k : v;
        const float scale = (part == 0) ? QK_SCALE : 1.0f;
#pragma unroll
        for (int r = 0; r < 8; ++r) {
            const int m = tm * 16 + r + half * 8;
            dst[(size_t)m * DIM + c] = (acc[t][r] + bias) * scale;
        }
    }
}

// ---------------------------------------------------------------------------
// Neighborhood attention: block = 1 point (8 waves), wave = 1 head.
// Phase 1: lane j=lane&15 owns neighbor j, half=lane>>4 owns 16 of the 32 dims;
//          shfl_xor(16) completes the dot, shfl_xor{8,4,2,1} does softmax stats.
// Phase 2: lane = output dim d; broadcast (w_j, nb_j) via shfl; coalesced v reads.
// ---------------------------------------------------------------------------
__global__ __launch_bounds__(256) void attn_kernel(
    const float* __restrict__ q,
    const float* __restrict__ k,
    const float* __restrict__ v,
    const int*   __restrict__ index_1,   // (N*K,)
    float* __restrict__ x)               // (N, 256)
{
    const int i    = blockIdx.x;
    const int h    = threadIdx.x >> 5;   // head 0..7
    const int lane = threadIdx.x & 31;
    const int j    = lane & 15;
    const int half = lane >> 4;

    const int nb = index_1[i * KNBR + j];

    const float* __restrict__ qrow = q + (size_t)i * DIM + h * HD + half * 16;
    const float* __restrict__ krow = k + (size_t)nb * DIM + h * HD + half * 16;

    float s = 0.f;
#pragma unroll
    for (int d = 0; d < 16; ++d) s += qrow[d] * krow[d];
    s += __shfl_xor(s, 16);              // full 32-dim dot, replicated in halves

    // softmax over the 16 neighbors (xor masks < 16 stay within each half)
    float mx = s;
    mx = fmaxf(mx, __shfl_xor(mx, 8));
    mx = fmaxf(mx, __shfl_xor(mx, 4));
    mx = fmaxf(mx, __shfl_xor(mx, 2));
    mx = fmaxf(mx, __shfl_xor(mx, 1));
    float e = __expf(s - mx);
    float sum = e;
    sum += __shfl_xor(sum, 8);
    sum += __shfl_xor(sum, 4);
    sum += __shfl_xor(sum, 2);
    sum += __shfl_xor(sum, 1);
    const float w = e / sum;

    // weighted sum of v rows; lane -> output dim
    float acc = 0.f;
#pragma unroll
    for (int jj = 0; jj < KNBR; ++jj) {
        const float wj  = __shfl(w, jj);
        const int   nbj = __shfl(nb, jj);
        acc += wj * v[(size_t)nbj * DIM + h * HD + lane];
    }
    x[(size_t)i * DIM + h * HD + lane] = acc;
}

// ---------------------------------------------------------------------------
// Output projection: out = x @ proj_w^T + proj_b, same 1x4-blocked WMMA tiling.
// ---------------------------------------------------------------------------
__global__ __launch_bounds__(128) void proj_kernel(
    const float* __restrict__ x,
    const float* __restrict__ proj_w,   // (256, 256)
    const float* __restrict__ proj_b,   // (256,)
    float* __restrict__ out)            // (N, 256)
{
    const int GROUPS_N = (DIM / 16) / NBLK;         // 4 groups of 4 col-tiles
    const int wave = (blockIdx.x * blockDim.x + threadIdx.x) >> 5;
    const int lane = threadIdx.x & 31;
    const int tm = wave / GROUPS_N;
    const int tn0 = (wave % GROUPS_N) * NBLK;
    const int row  = lane & 15;
    const int half = lane >> 4;

    const float* __restrict__ arow = x + (size_t)(tm * 16 + row) * DIM + half * 2;
    const float* __restrict__ brow[NBLK];
#pragma unroll
    for (int t = 0; t < NBLK; ++t)
        brow[t] = proj_w + (size_t)((tn0 + t) * 16 + row) * DIM + half * 2;

    v8f acc[NBLK];
#pragma unroll
    for (int t = 0; t < NBLK; ++t) acc[t] = (v8f){0.f,0.f,0.f,0.f,0.f,0.f,0.f,0.f};

#pragma unroll 4
    for (int k0 = 0; k0 < DIM; k0 += 4) {
        v2f a = *(const v2f*)(arow + k0);
#pragma unroll
        for (int t = 0; t < NBLK; ++t) {
            v2f b = *(const v2f*)(brow[t] + k0);
            acc[t] = wmma_f32_16x16x4(a, b, acc[t]);
        }
    }

#pragma unroll
    for (int t = 0; t < NBLK; ++t) {
        const int col = (tn0 + t) * 16 + row;
        const float bias = proj_b[col];
#pragma unroll
        for (int r = 0; r < 8; ++r) {
            const int m = tm * 16 + r + half * 8;
            out[(size_t)m * DIM + col] = acc[t][r] + bias;
        }
    }
}

// ---------------------------------------------------------------------------
extern "C" void kernel_launch(void* const* d_in, const int* in_sizes, int n_in,
                              void* d_out, int out_size, void* d_ws, size_t ws_size,
                              hipStream_t stream) {
    const float* feats   = (const float*)d_in[0];
    // d_in[1] xyz            — unused by the reference math
    // d_in[2] index_0        — implicit (i = m / K)
    const int*   index_1 = (const int*)d_in[3];
    // d_in[4] index_0_offsets, d_in[5] n_max — fixed K=16 layout
    const float* qkv_w   = (const float*)d_in[6];
    const float* qkv_b   = (const float*)d_in[7];
    const float* proj_w  = (const float*)d_in[8];
    const float* proj_b  = (const float*)d_in[9];
    float* out = (float*)d_out;

    const size_t plane = (size_t)NPTS * DIM;        // 12.8M floats
    float* q  = (float*)d_ws;
    float* k  = q + plane;
    float* v  = k + plane;
    float* x  = v + plane;                          // total 204.8 MB

    // QKV GEMM: 3125 M-tiles x 12 col-groups = 37500 waves, 4 waves/block
    qkv_kernel<<<9375, 128, 0, stream>>>(feats, qkv_w, qkv_b, q, k, v);

    // Attention: one block per point
    attn_kernel<<<NPTS, 256, 0, stream>>>(q, k, v, index_1, x);

    // Projection GEMM: 3125 M-tiles x 4 col-groups = 12500 waves
    proj_kernel<<<3125, 128, 0, stream>>>(x, proj_w, proj_b, out);
}